// SpikeSEGEncoder_21672404975776
// MI455X (gfx1250) — compile-verified
//
#include <hip/hip_runtime.h>
#include <math.h>
#include <stdint.h>

// SNN encoder: 3x (conv -> LIF[-WTA] [-> maxpool]) over T=20 timesteps.
// Convs are implicit-im2col GEMMs on V_WMMA_F32_16X16X4_F32 (full-fp32 WMMA,
// preserves reference numerics). Weights are staged to LDS via the CDNA5
// async global->LDS path and consumed with ds_load_b64. K loops are fully
// unrolled for conv1/conv2 so all im2col decode is compile-time.

typedef __attribute__((ext_vector_type(2))) float v2f;
typedef __attribute__((ext_vector_type(8))) float v8f;

#define THRV 10.0f

// Issue one CDNA5 async global->LDS dword copy (tracked by ASYNCcnt).
__device__ __forceinline__ void async_g2l_b32(uint32_t lds_off, const float* g) {
  asm volatile("global_load_async_to_lds_b32 %0, %1, off"
               :: "v"(lds_off), "v"((uint64_t)(uintptr_t)g)
               : "memory");
}
__device__ __forceinline__ void wait_asynccnt0() {
  asm volatile("s_wait_asynccnt 0" ::: "memory");
}
// Generic shared pointer -> wave-relative LDS byte offset (low 32 bits of the
// flat shared-aperture address are the LDS offset on gfx1250).
__device__ __forceinline__ uint32_t lds_off_of(const void* p) {
  return (uint32_t)(uintptr_t)p;
}

// ---------------------------------------------------------------------------
// Implicit-im2col conv as WMMA GEMM.
//   in  : (B, CIN, H, W)   wgt : (COUT, CIN, KH, KW)   out : (B, COUT, H, W)
// GEMM view: M = B*H*W output pixels, N = COUT, K = CIN*KH*KW (SAME padding).
// One wave computes a 16(M) x 16(N) f32 tile, K consumed in chunks of 4.
// f32 WMMA fragment layout (ISA 7.12.2):
//   A (16x4): lanes 0-15 -> M=0..15, v0=K0,v1=K1; lanes 16-31 -> same M, K2/K3
//   B (4x16): mirrored over N;  C/D: VGPR r -> M=r+8*(lane>=16), N=lane%16
// ---------------------------------------------------------------------------
template<int CIN, int KH, int KW, int H, int W, int COUT>
__global__ __launch_bounds__(128)
void conv_wmma_kernel(const float* __restrict__ in, const float* __restrict__ wgt,
                      float* __restrict__ out, int B) {
  constexpr int KTOT = CIN * KH * KW;
  constexpr int KCH  = (KTOT + 3) / 4;          // K chunks of 4
  constexpr int KPAD = KCH * 4;                 // LDS row stride (floats)
  constexpr int PH = KH / 2, PW = KW / 2;       // SAME padding (odd kernels)
  constexpr int ROWS = (COUT < 16) ? COUT : 16; // weight rows resident in LDS

  __shared__ float wlds[ROWS][KPAD];

  const int tid   = threadIdx.y * 32 + threadIdx.x;
  const int nBase = blockIdx.y * 16;

  // ---- stage this N-tile's weights into LDS ------------------------------
  // Zero only the padded K columns (they multiply into *valid* D columns).
  if constexpr (KPAD != KTOT) {
    constexpr int TAIL = KPAD - KTOT;
    for (int i = tid; i < ROWS * TAIL; i += 128)
      wlds[i / TAIL][KTOT + (i % TAIL)] = 0.0f;
  }
  // Async-copy the valid weight block; rows with n >= COUT stay garbage but
  // only feed D columns that are never stored (WMMA columns are independent).
  for (int i = tid; i < ROWS * KTOT; i += 128) {
    const int r = i / KTOT, k = i % KTOT;
    const int n = nBase + r;
    if (n < COUT)
      async_g2l_b32(lds_off_of(&wlds[r][k]), wgt + (size_t)n * KTOT + k);
  }
  wait_asynccnt0();
  __syncthreads();

  // ---- per-wave GEMM tile -------------------------------------------------
  const int lane = threadIdx.x;     // 0..31
  const int hf   = lane >> 4;       // wave half selects K pair {0,1}/{2,3}
  const int l    = lane & 15;

  const int mTile = blockIdx.x * blockDim.y + threadIdx.y;
  const int mBase = mTile * 16;
  const int NP    = H * W;
  const int total = B * NP;
  if (mBase >= total) return;       // wave-uniform (EXEC stays all-1 for WMMA)

  const int  mIdx   = mBase + l;
  const bool mValid = (mIdx < total);
  int b = 0, y = 0, x = 0;
  if (mValid) { b = mIdx / NP; int r = mIdx % NP; y = r / W; x = r % W; }
  const float* aBase = in + (size_t)b * CIN * NP + y * W + x;

  // SAME-padding validity as bitmasks: A element needs ymask[kh] & xmask[kw].
  int ymask = 0, xmask = 0;
#pragma unroll
  for (int i = 0; i < KH; ++i) {
    const int iy = y + i - PH;
    if (mValid && iy >= 0 && iy < H) ymask |= (1 << i);
  }
#pragma unroll
  for (int i = 0; i < KW; ++i) {
    const int ix = x + i - PW;
    if (ix >= 0 && ix < W) xmask |= (1 << i);
  }

  const int  brow   = (l < ROWS) ? l : (ROWS - 1); // clamp LDS row (safe read)
  const int  nOut   = nBase + l;
  const bool nValid = (nOut < COUT);

  auto aElem = [&](int k) -> float {              // im2col gather for index k
    if (k >= KTOT) return 0.0f;
    const int ci = k / (KH * KW);
    const int rr = k % (KH * KW);
    const int kh = rr / KW, kw = rr % KW;
    float av = 0.0f;
    if (((ymask >> kh) & 1) && ((xmask >> kw) & 1))
      av = aBase[ci * NP + (kh - PH) * W + (kw - PW)];
    return av;
  };

  v8f acc = {};
  auto kchunk = [&](int kc) {
    const int k0 = kc * 4 + hf * 2;               // even -> 8B-aligned LDS pair
    v2f a;
    a[0] = aElem(k0);
    a[1] = aElem(k0 + 1);
    const v2f bfrag = *(const v2f*)&wlds[brow][k0];   // ds_load_b64
    acc = __builtin_amdgcn_wmma_f32_16x16x4_f32(
        false, a, false, bfrag, (short)0, acc, false, false);
  };

  if constexpr (KCH <= 32) {
#pragma unroll
    for (int kc = 0; kc < KCH; ++kc) kchunk(kc);  // fully compile-time decode
  } else {
    for (int kc = 0; kc < KCH; ++kc) kchunk(kc);  // conv3 (K=1764)
  }

#pragma unroll
  for (int r = 0; r < 8; ++r) {
    const int M = mBase + r + hf * 8;
    if (M < total && nValid) {
      const int b2 = M / NP; const int rem = M % NP;
      const int y2 = rem / W; const int x2 = rem % W;
      out[((size_t)(b2 * COUT + nOut) * H + y2) * W + x2] = acc[r];
    }
  }
}

// ---------------------------------------------------------------------------
// Fused LIF + winner-take-all ("both" mode). Each thread owns one 2x2 spatial
// block across ALL C channels of one batch image -> channel argmax and local
// spatial max are both thread-local. io holds the conv input on entry and the
// spike output on exit (in-place); m is the membrane state (updated).
// ---------------------------------------------------------------------------
template<int C>
__global__ void lif_wta_kernel(float* __restrict__ m, float* __restrict__ io,
                               float leak, int B, int H, int W) {
  const int HB = H / 2, WB = W / 2;
  const int total = B * HB * WB;
  const int idx = blockIdx.x * blockDim.x + threadIdx.x;
  if (idx >= total) return;
  const int b  = idx / (HB * WB);
  const int rm = idx % (HB * WB);
  const int by = (rm / WB) * 2, bx = (rm % WB) * 2;

  // Pass 1: per-position channel winner (first max wins, like jnp.argmax).
  float best[4] = {-INFINITY, -INFINITY, -INFINITY, -INFINITY};
  int   bi[4]   = {0, 0, 0, 0};
  for (int c = 0; c < C; ++c) {
    const size_t base = ((size_t)(b * C + c) * H + by) * W + bx;
#pragma unroll
    for (int p = 0; p < 4; ++p) {
      const int off = (p >> 1) * W + (p & 1);
      const float v = m[base + off] + io[base + off];
      if (v > best[p]) { best[p] = v; bi[p] = c; }
    }
  }
  // Pass 2: spikes + reset-to-zero + subtractive leak.
  for (int c = 0; c < C; ++c) {
    const size_t base = ((size_t)(b * C + c) * H + by) * W + bx;
    float v[4];
    float bmax = -INFINITY;
#pragma unroll
    for (int p = 0; p < 4; ++p) {
      const int off = (p >> 1) * W + (p & 1);
      v[p] = m[base + off] + io[base + off];
      bmax = fmaxf(bmax, v[p]);
    }
#pragma unroll
    for (int p = 0; p < 4; ++p) {
      const int off = (p >> 1) * W + (p & 1);
      float mem = v[p];
      float spk = (mem >= THRV) ? 1.0f : 0.0f;
      if (bi[p] != c)  spk = 0.0f;   // channel competition
      if (mem < bmax)  spk = 0.0f;   // local spatial competition (ties pass)
      float nm = (spk > 0.0f) ? 0.0f : mem;
      nm = fmaxf(nm - leak, 0.0f);
      m[base + off]  = nm;
      io[base + off] = spk;
    }
  }
}

// Layer-3 LIF (no WTA); writes spikes straight into d_out slice.
__global__ void lif3_kernel(float* __restrict__ m, const float* __restrict__ inp,
                            float* __restrict__ spkOut, float leak, int n) {
  const int i = blockIdx.x * blockDim.x + threadIdx.x;
  if (i >= n) return;
  float mem = m[i] + inp[i];
  const float spk = (mem >= THRV) ? 1.0f : 0.0f;
  mem = (spk > 0.0f) ? 0.0f : mem;
  mem = fmaxf(mem - leak, 0.0f);
  m[i] = mem;
  spkOut[i] = spk;
}

// VALID max-pool, one thread per output element.
__global__ void pool_kernel(const float* __restrict__ in, float* __restrict__ out,
                            int BC, int inH, int inW, int outH, int outW,
                            int k, int s) {
  const int total = BC * outH * outW;
  const int idx = blockIdx.x * blockDim.x + threadIdx.x;
  if (idx >= total) return;
  const int bc = idx / (outH * outW);
  const int rm = idx % (outH * outW);
  const int oy = rm / outW, ox = rm % outW;
  float mv = -INFINITY;
  for (int dy = 0; dy < k; ++dy)
    for (int dx = 0; dx < k; ++dx)
      mv = fmaxf(mv, in[((size_t)bc * inH + oy * s + dy) * inW + ox * s + dx]);
  out[idx] = mv;
}

__global__ void zero_kernel(float* __restrict__ p, int n) {
  const int i = blockIdx.x * blockDim.x + threadIdx.x;
  if (i < n) p[i] = 0.0f;
}

// ---------------------------------------------------------------------------
extern "C" void kernel_launch(void* const* d_in, const int* in_sizes, int n_in,
                              void* d_out, int out_size, void* d_ws, size_t ws_size,
                              hipStream_t stream) {
  (void)in_sizes; (void)n_in; (void)out_size; (void)ws_size;
  const float* x  = (const float*)d_in[0];  // (20,8,1,256,256)
  const float* w1 = (const float*)d_in[1];  // (4,1,5,5)
  const float* w2 = (const float*)d_in[2];  // (36,4,5,5)
  const float* w3 = (const float*)d_in[3];  // (1,36,7,7)
  float* out = (float*)d_out;               // (20,8,1,21,21)

  constexpr int T = 20, B = 8;
  constexpr int H1 = 256, W1 = 256, C1 = 4;
  constexpr int H2 = 42,  W2 = 42,  C2 = 36;
  constexpr int H3 = 21,  W3 = 21,  C3 = 1;

  float* ws = (float*)d_ws;
  size_t o = 0;
  float* m1   = ws + o; o += (size_t)B * C1 * H1 * W1;  // membrane L1
  float* buf1 = ws + o; o += (size_t)B * C1 * H1 * W1;  // conv1 out -> s1
  float* p1   = ws + o; o += (size_t)B * C1 * H2 * W2;  // pooled s1
  float* m2   = ws + o; o += (size_t)B * C2 * H2 * W2;  // membrane L2
  float* buf2 = ws + o; o += (size_t)B * C2 * H2 * W2;  // conv2 out -> s2
  float* p2   = ws + o; o += (size_t)B * C2 * H3 * W3;  // pooled s2
  float* m3   = ws + o; o += (size_t)B * C3 * H3 * W3;  // membrane L3
  float* buf3 = ws + o; o += (size_t)B * C3 * H3 * W3;  // conv3 out
  const int wsFloats = (int)o;                          // ~5.4M floats (21.6 MB)

  // Deterministic start: zero membranes (and scratch) every call.
  zero_kernel<<<(wsFloats + 255) / 256, 256, 0, stream>>>(ws, wsFloats);

  for (int t = 0; t < T; ++t) {
    const float* xt = x + (size_t)t * B * 1 * H1 * W1;

    { // conv1: 1->4 ch, 5x5 SAME (K=25 padded to 28, fully unrolled)
      const int tiles = (B * H1 * W1 + 15) / 16;        // 32768 M-tiles
      conv_wmma_kernel<1, 5, 5, H1, W1, C1>
          <<<dim3((tiles + 3) / 4, 1), dim3(32, 4), 0, stream>>>(xt, w1, buf1, B);
    }
    { // LIF1 + WTA (leak 9)
      const int n = B * (H1 / 2) * (W1 / 2);
      lif_wta_kernel<C1><<<(n + 255) / 256, 256, 0, stream>>>(m1, buf1, 9.0f, B, H1, W1);
    }
    { // pool 7/6 : 256 -> 42
      const int n = B * C1 * H2 * W2;
      pool_kernel<<<(n + 255) / 256, 256, 0, stream>>>(buf1, p1, B * C1, H1, W1, H2, W2, 7, 6);
    }
    { // conv2: 4->36 ch, 5x5 SAME (K=100, fully unrolled)
      const int tiles = (B * H2 * W2 + 15) / 16;        // 882 M-tiles
      conv_wmma_kernel<4, 5, 5, H2, W2, C2>
          <<<dim3((tiles + 3) / 4, (C2 + 15) / 16), dim3(32, 4), 0, stream>>>(p1, w2, buf2, B);
    }
    { // LIF2 + WTA (leak 1)
      const int n = B * (H2 / 2) * (W2 / 2);
      lif_wta_kernel<C2><<<(n + 255) / 256, 256, 0, stream>>>(m2, buf2, 1.0f, B, H2, W2);
    }
    { // pool 2/2 : 42 -> 21
      const int n = B * C2 * H3 * W3;
      pool_kernel<<<(n + 255) / 256, 256, 0, stream>>>(buf2, p2, B * C2, H2, W2, H3, W3, 2, 2);
    }
    { // conv3: 36->1 ch, 7x7 SAME (K=1764, runtime K loop)
      const int tiles = (B * H3 * W3 + 15) / 16;        // 221 M-tiles
      conv_wmma_kernel<36, 7, 7, H3, W3, C3>
          <<<dim3((tiles + 3) / 4, 1), dim3(32, 4), 0, stream>>>(p2, w3, buf3, B);
    }
    { // LIF3 (no WTA, leak 0) -> output spikes slice t
      const int n = B * C3 * H3 * W3;
      lif3_kernel<<<(n + 255) / 256, 256, 0, stream>>>(m3, buf3, out + (size_t)t * n, 0.0f, n);
    }
  }
}